// SMPLModel_76115410420047
// MI455X (gfx1250) — compile-verified
//
#include <hip/hip_runtime.h>
#include <math.h>

#define P_   200
#define N_   6890
#define K_   24
#define KJ   23
#define PAD_ 17
#define M3   (3 * N_)    /* 20670 */
#define PPAD 208         /* 13*16 padded poses */
#define RPAD 20          /* 17 padded to 5 k-steps of 4 */
#define MT   1292        /* ceil(20670/16) */
#define PT   13          /* ceil(200/16)   */

typedef __attribute__((ext_vector_type(2))) float v2f;
typedef __attribute__((ext_vector_type(8))) float v8f;

// ---- workspace layout (bytes) ----
#define OFF_ACC 0                                      // 27 doubles
#define OFF_G   512                                    // P*K*16 f32
#define OFF_QB  (OFF_G  + (size_t)P_ * K_ * 16 * 4)    // P*K*4 f32
#define OFF_QT  (OFF_QB + (size_t)P_ * K_ * 4 * 4)     // KJ*RPAD*PPAD f32
#define OFF_TP  (OFF_QT + (size_t)KJ * RPAD * PPAD * 4)// PPAD*M3 f32
#define OFF_KP  (OFF_TP + (size_t)PPAD * M3 * 4)       // KJ*4*M3 f32 (rows 16..19, zero-padded)

__constant__ int c_nbr[24][4] = {
  {0,1,2,3},{0,1,4,-1},{0,2,5,-1},{0,3,6,-1},{1,4,7,-1},{2,5,8,-1},
  {3,6,9,-1},{4,7,10,-1},{5,8,11,-1},{6,9,-1,-1},{7,10,-1,-1},{8,11,-1,-1},
  {12,13,14,15},{12,13,16,-1},{12,14,17,-1},{12,15,-1,-1},{13,16,18,-1},
  {14,17,19,-1},{16,18,20,-1},{17,19,21,-1},{18,20,22,-1},{19,21,23,-1},
  {20,22,-1,-1},{21,23,-1,-1}};
__constant__ int c_cnt[24] = {4,3,3,3,3,3,3,3,3,2,2,2,4,3,3,2,3,3,3,3,3,3,2,2};

__device__ __forceinline__ float blockReduce(float v, float* sm) {
  int lane = threadIdx.x & 31, wid = threadIdx.x >> 5;
  #pragma unroll
  for (int o = 16; o > 0; o >>= 1) v += __shfl_down(v, o, 32);
  if (lane == 0) sm[wid] = v;
  __syncthreads();
  float r = 0.f;
  if (threadIdx.x < (blockDim.x >> 5)) r = sm[threadIdx.x];
  if (wid == 0) {
    #pragma unroll
    for (int o = 16; o > 0; o >>= 1) r += __shfl_down(r, o, 32);
  }
  __syncthreads();  // allow helper reuse
  return r;         // valid in thread 0
}

// ---- K0: zero accumulators ----
__global__ void k_zero(double* acc) {
  int i = threadIdx.x;
  if (i < 27) acc[i] = 0.0;
}

// ---- K1: per (p,k) quaternion (minus identity) + Rodrigues G ----
__global__ void k_pose(const float* __restrict__ theta, const float* __restrict__ J,
                       float* __restrict__ G, float* __restrict__ qb) {
  int idx = blockIdx.x * blockDim.x + threadIdx.x;
  if (idx >= P_ * K_) return;
  float tx = theta[idx*3+0], ty = theta[idx*3+1], tz = theta[idx*3+2];
  float hx = 0.5f*tx, hy = 0.5f*ty, hz = 0.5f*tz;
  float nn = sqrtf(hx*hx + hy*hy + hz*hz);
  float sn = sinf(nn) / (nn + 1e-8f);
  qb[idx*4+0] = cosf(nn) - 1.0f;
  qb[idx*4+1] = hx*sn; qb[idx*4+2] = hy*sn; qb[idx*4+3] = hz*sn;
  float tn  = sqrtf(tx*tx + ty*ty + tz*tz);
  float inv = 1.0f / (tn + 1e-8f);
  float x = tx*inv, y = ty*inv, z = tz*inv;
  float s = sinf(tn), cc = 1.0f - cosf(tn);
  float* g = G + (size_t)idx * 16;
  g[0]  = 1.f + cc*(-(y*y + z*z)); g[1]  = -s*z + cc*(x*y); g[2]  =  s*y + cc*(x*z);
  g[4]  =  s*z + cc*(x*y);         g[5]  = 1.f + cc*(-(x*x + z*z)); g[6] = -s*x + cc*(y*z);
  g[8]  = -s*y + cc*(x*z);         g[9]  =  s*x + cc*(y*z); g[10] = 1.f + cc*(-(x*x + y*y));
  g[3]  = J[idx*3+0]; g[7] = J[idx*3+1]; g[11] = J[idx*3+2];
  g[12] = 0.f; g[13] = 0.f; g[14] = 0.f; g[15] = 1.f;
}

// ---- K2: build q_ne transposed + zero-padded: qT[j][r<20][p<208] ----
__global__ void k_qne(const float* __restrict__ qb, const float* __restrict__ beta2,
                      float* __restrict__ qT) {
  int idx = blockIdx.x * blockDim.x + threadIdx.x;
  if (idx >= PPAD * KJ) return;
  int p = idx % PPAD, j = idx / PPAD;
  float row[RPAD];
  #pragma unroll
  for (int r = 0; r < RPAD; ++r) row[r] = 0.f;
  if (p < P_) {
    int jj = j + 1, cnt = c_cnt[jj];
    for (int t = 0; t < cnt; ++t) {
      int nk = c_nbr[jj][t];
      const float* q = qb + ((size_t)p * K_ + nk) * 4;
      row[4*t+0] = q[0]; row[4*t+1] = q[1]; row[4*t+2] = q[2]; row[4*t+3] = q[3];
    }
    row[4*cnt] = beta2[0];
  }
  for (int r = 0; r < RPAD; ++r) qT[((size_t)j * RPAD + r) * PPAD + p] = row[r];
}

// ---- K2b: zero-padded K tail rows 16..19: Kpad[j][rr][m] ----
__global__ void k_kpad(const float* __restrict__ Kp, float* __restrict__ Kpad) {
  int idx = blockIdx.x * blockDim.x + threadIdx.x;
  const int total = KJ * 4 * M3;
  if (idx >= total) return;
  int m = idx % M3;
  int t = idx / M3;
  int rr = t % 4, j = t / 4;
  Kpad[idx] = (rr == 0) ? Kp[((size_t)j * PAD_ + 16) * M3 + m] : 0.f;
}

// ---- K3: single-chain WMMA f32 16x16x4 GEMM with relu(A) folded into B ----
// Tp[p][m] = sum_{j,r} qT[j][r][p] * ( relu(A[j, m/3]) * Kp[j][r][m] )
__global__ void __launch_bounds__(256) k_gemm(const float* __restrict__ qT,
                                              const float* __restrict__ Kp,
                                              const float* __restrict__ Kpad,
                                              const float* __restrict__ A,
                                              float* __restrict__ Tp) {
  int wave = (blockIdx.x * blockDim.x + threadIdx.x) >> 5;
  if (wave >= PT * MT) return;          // uniform per wave -> EXEC all-ones for WMMA
  int lane = threadIdx.x & 31;
  int tm = wave % MT, tpi = wave / MT;
  int m0 = tm * 16, p0 = tpi * 16;
  int col = lane & 15, hi = lane >> 4;
  int m = m0 + col;
  bool mv = (m < M3);
  int mi = mv ? m : (M3 - 1);           // clamp: OOB lanes compute garbage in their
  int nv = mi / 3;                      // own column only; store is predicated
  v8f acc = {};
  for (int j = 0; j < KJ; ++j) {
    float scale = fmaxf(A[(size_t)j * N_ + nv], 0.f);
    const float* qTj = qT   + (size_t)j * RPAD * PPAD + p0 + col;
    const float* Kpj = Kp   + (size_t)j * PAD_ * M3 + mi;
    const float* Kqj = Kpad + (size_t)j * 4    * M3 + mi;
    #pragma unroll
    for (int s = 0; s < 5; ++s) {
      int r0 = 4 * s + 2 * hi;
      v2f a, b;
      a.x = qTj[(size_t)r0 * PPAD];           // zero-padded for r>=17, p>=200
      a.y = qTj[(size_t)(r0 + 1) * PPAD];
      if (s < 4) {
        b.x = Kpj[(size_t)r0 * M3];           // r0 <= 14, r0+1 <= 15 : in-bounds
        b.y = Kpj[(size_t)(r0 + 1) * M3];
      } else {
        b.x = Kqj[(size_t)(2 * hi)     * M3]; // rows 16..19 via zero-padded tail
        b.y = Kqj[(size_t)(2 * hi + 1) * M3];
      }
      b.x *= scale;
      b.y *= scale;
      acc = __builtin_amdgcn_wmma_f32_16x16x4_f32(false, a, false, b,
                                                  (short)0, acc, false, false);
    }
  }
  if (mv) {
    #pragma unroll
    for (int i = 0; i < 8; ++i)
      Tp[(size_t)(p0 + i + 8 * hi) * M3 + m] = acc[i];
  }
}

// ---- K4: per-vertex skinning + tv + E_D partial ----
__global__ void __launch_bounds__(256) k_vertex(const float* __restrict__ Tp,
    const float* __restrict__ T, const float* __restrict__ V,
    const float* __restrict__ W, const float* __restrict__ G,
    float* __restrict__ tv, double* __restrict__ acc) {
  __shared__ float Gs[K_ * 16];
  __shared__ float sm[8];
  int p = blockIdx.y, tid = threadIdx.x;
  for (int i = tid; i < K_ * 16; i += blockDim.x) Gs[i] = G[(size_t)p * K_ * 16 + i];
  __syncthreads();
  int n = blockIdx.x * blockDim.x + tid;
  float ed = 0.f;
  if (n < N_) {
    size_t base = ((size_t)p * N_ + n) * 3;
    size_t tb = (size_t)p * M3 + 3 * (size_t)n;
    float th0 = T[base+0] + Tp[tb+0];
    float th1 = T[base+1] + Tp[tb+1];
    float th2 = T[base+2] + Tp[tb+2];
    float a0 = 0.f, a1 = 0.f, a2 = 0.f;
    #pragma unroll
    for (int k = 0; k < K_; ++k) {
      float w = W[(size_t)n * K_ + k];
      const float* g = &Gs[k * 16];
      a0 += w * (g[0]*th0 + g[1]*th1 + g[2]*th2  + g[3]);
      a1 += w * (g[4]*th0 + g[5]*th1 + g[6]*th2  + g[7]);
      a2 += w * (g[8]*th0 + g[9]*th1 + g[10]*th2 + g[11]);
    }
    tv[base+0] = a0; tv[base+1] = a1; tv[base+2] = a2;
    float d0 = V[base+0]-a0, d1 = V[base+1]-a1, d2 = V[base+2]-a2;
    ed = d0*d0 + d1*d1 + d2*d2;
  }
  float r = blockReduce(ed, sm);
  if (tid == 0) atomicAdd(&acc[0], (double)r);
}

// ---- K5: E_Wi, E_W ----
__global__ void k_wterms(const float* __restrict__ W, const float* __restrict__ Wi,
                         double* acc) {
  __shared__ float sm[8];
  float a = 0.f, b = 0.f;
  int total = N_ * K_;
  for (int i = blockIdx.x * blockDim.x + threadIdx.x; i < total;
       i += gridDim.x * blockDim.x) {
    float w = W[i], d = w - Wi[i];
    a += d * d; b += fabsf(w);
  }
  float ra = blockReduce(a, sm);
  float rb = blockReduce(b, sm);
  if (threadIdx.x == 0) { atomicAdd(&acc[1], (double)ra); atomicAdd(&acc[2], (double)rb); }
}

// ---- K6: E_A ----
__global__ void k_aterm(const float* __restrict__ A, double* acc) {
  __shared__ float sm[8];
  float a = 0.f;
  int total = KJ * N_;
  for (int i = blockIdx.x * blockDim.x + threadIdx.x; i < total;
       i += gridDim.x * blockDim.x) a += fabsf(A[i]);
  float ra = blockReduce(a, sm);
  if (threadIdx.x == 0) atomicAdd(&acc[3], (double)ra);
}

// ---- K7: per-j sum of K_params^2 ----
__global__ void k_ek(const float* __restrict__ Kp, double* acc) {
  __shared__ float sm[8];
  int j = blockIdx.y;
  const float* kp = Kp + (size_t)j * PAD_ * M3;
  float s = 0.f;
  int total = PAD_ * M3;
  for (int i = blockIdx.x * blockDim.x + threadIdx.x; i < total;
       i += gridDim.x * blockDim.x) { float v = kp[i]; s += v * v; }
  float r = blockReduce(s, sm);
  if (threadIdx.x == 0) atomicAdd(&acc[4 + j], (double)r);
}

// ---- K8: combine energies ----
__global__ void k_final(const double* __restrict__ acc, float* __restrict__ out) {
  if (threadIdx.x == 0 && blockIdx.x == 0) {
    double E = 2.5 * acc[0] + 0.1 * acc[1] + 5.0 * acc[2] + 2.5 * acc[3];
    double ek = 0.0;
    for (int j = 0; j < KJ; ++j) ek += sqrt(acc[4 + j]);
    out[0] = (float)(E + 2.5 * ek);
  }
}

extern "C" void kernel_launch(void* const* d_in, const int* in_sizes, int n_in,
                              void* d_out, int out_size, void* d_ws, size_t ws_size,
                              hipStream_t stream) {
  (void)in_sizes; (void)n_in; (void)out_size; (void)ws_size;
  const float* V     = (const float*)d_in[0];
  const float* T     = (const float*)d_in[1];
  const float* J     = (const float*)d_in[2];
  const float* theta = (const float*)d_in[3];
  const float* beta2 = (const float*)d_in[4];
  const float* W     = (const float*)d_in[5];
  const float* Wi    = (const float*)d_in[6];
  const float* A     = (const float*)d_in[7];
  const float* Kp    = (const float*)d_in[8];

  char* ws    = (char*)d_ws;
  double* acc = (double*)(ws + OFF_ACC);
  float* G    = (float*)(ws + OFF_G);
  float* qb   = (float*)(ws + OFF_QB);
  float* qT   = (float*)(ws + OFF_QT);
  float* Tp   = (float*)(ws + OFF_TP);
  float* Kpad = (float*)(ws + OFF_KP);
  float* out  = (float*)d_out;

  k_zero<<<1, 32, 0, stream>>>(acc);
  k_pose<<<(P_ * K_ + 255) / 256, 256, 0, stream>>>(theta, J, G, qb);
  k_qne<<<(PPAD * KJ + 255) / 256, 256, 0, stream>>>(qb, beta2, qT);
  k_kpad<<<(KJ * 4 * M3 + 255) / 256, 256, 0, stream>>>(Kp, Kpad);
  k_gemm<<<(PT * MT * 32 + 255) / 256, 256, 0, stream>>>(qT, Kp, Kpad, A, Tp);
  k_vertex<<<dim3((N_ + 255) / 256, P_), 256, 0, stream>>>(Tp, T, V, W, G, out + 1, acc);
  k_wterms<<<128, 256, 0, stream>>>(W, Wi, acc);
  k_aterm<<<128, 256, 0, stream>>>(A, acc);
  k_ek<<<dim3(86, KJ), 256, 0, stream>>>(Kp, acc);
  k_final<<<1, 1, 0, stream>>>(acc, out);
}